// ResidualModelPassMessage_76192719831683
// MI455X (gfx1250) — compile-verified
//
#include <hip/hip_runtime.h>
#include <hip/hip_bf16.h>
#include <math.h>

typedef float v2f __attribute__((ext_vector_type(2)));
typedef float v8f __attribute__((ext_vector_type(8)));

#define N_NODES 100000
#define N_EDGES 1600000
#define FEATS   128
// concat dim = 128 + 128 + 1 = 257

// ---------------------------------------------------------------- fill zero
__global__ void fill0_kernel(float4* __restrict__ p, int n4) {
    int i = blockIdx.x * blockDim.x + threadIdx.x;
    if (i < n4) p[i] = make_float4(0.f, 0.f, 0.f, 0.f);
}

// ------------------------------------------------- edge-feature segment sum (done once)
__global__ void edge_aggr_e_kernel(const float* __restrict__ he,
                                   const int* __restrict__ dst,
                                   float* __restrict__ he_aggr) {
    int e = blockIdx.x * blockDim.x + threadIdx.x;
    if (e < N_EDGES) atomicAdd(&he_aggr[dst[e]], he[e]);
}

// ------------------------------------------------- node-feature segment sum (per layer)
// One wave (32 lanes) per edge: lane L moves 4 contiguous floats (float4 load,
// 4x global_atomic_add_f32). Feature matrices are L2-resident (51 MB < 192 MB L2).
__global__ void __launch_bounds__(256)
edge_aggr_feat_kernel(const float* __restrict__ h,
                      const int* __restrict__ src,
                      const int* __restrict__ dst,
                      float* __restrict__ hagg) {
    int gtid = blockIdx.x * blockDim.x + threadIdx.x;
    int e    = gtid >> 5;
    int lane = gtid & 31;
    if (e >= N_EDGES) return;
    int s = src[e];
    int d = dst[e];
    const float4 v = *reinterpret_cast<const float4*>(h + (size_t)s * FEATS + lane * 4);
    float* dp = hagg + (size_t)d * FEATS + lane * 4;
    atomicAdd(dp + 0, v.x);
    atomicAdd(dp + 1, v.y);
    atomicAdd(dp + 2, v.z);
    atomicAdd(dp + 3, v.w);
}

// ------------------------------------------------- WMMA GEMM (K=128 block accumulate)
// V_WMMA_F32_16X16X4_F32 layouts (ISA 7.12.2):
//  A 16x4: lanes 0-15 hold M=lane, K={0,1} in v0/v1; lanes 16-31 K={2,3}
//  B 4x16: lanes 0-15 hold N=lane, K={0,1} in v0/v1; lanes 16-31 K={2,3}
//  C/D 16x16: VGPR r -> M = r + 8*(lane>=16), N = lane&15
__device__ __forceinline__
v8f wmma_acc_k128(const float* __restrict__ X,   // [nodes][128], rows row0..row0+15
                  const float* __restrict__ W,   // [128][128] row-major slice of weights
                  int row0, int colbase, int laneM, int half, v8f c) {
    const float* xrow = X + (size_t)(row0 + laneM) * FEATS + 2 * half;
    const float* wcol = W + (size_t)(2 * half) * FEATS + colbase + laneM;
#pragma unroll 8
    for (int k = 0; k < 128; k += 4) {
        v2f a, b;
        a.x = xrow[k];
        a.y = xrow[k + 1];
        b.x = wcol[(size_t)k * FEATS];
        b.y = wcol[(size_t)k * FEATS + FEATS];
        c = __builtin_amdgcn_wmma_f32_16x16x4_f32(false, a, false, b,
                                                  (short)0, c, false, false);
    }
    return c;
}

// act: 0 = relu-residual (z + max(z,0)),  1 = sigmoid
__global__ void __launch_bounds__(256)
sage_gemm_wmma_kernel(const float* __restrict__ h,
                      const float* __restrict__ hagg,
                      const float* __restrict__ hea,
                      const float* __restrict__ W,     // [257][128]
                      const float* __restrict__ bias,  // [128]
                      float* __restrict__ out,         // [nodes][128]
                      int act) {
    const int row0    = blockIdx.x * 16;
    const int wave    = threadIdx.x >> 5;
    const int lane    = threadIdx.x & 31;
    const int half    = lane >> 4;
    const int laneM   = lane & 15;
    const int colbase = wave * 16;

    v8f c = {};
    c = wmma_acc_k128(h,    W,                          row0, colbase, laneM, half, c);
    c = wmma_acc_k128(hagg, W + (size_t)128 * FEATS,    row0, colbase, laneM, half, c);

    const int   col   = colbase + laneM;
    const float wlast = W[(size_t)256 * FEATS + col];
    const float bcol  = bias[col];
#pragma unroll
    for (int r = 0; r < 8; ++r) {
        int   row = row0 + r + 8 * half;
        float z   = c[r] + hea[row] * wlast + bcol;
        float o;
        if (act == 0) o = z + fmaxf(z, 0.0f);
        else          o = 1.0f / (1.0f + expf(-z));
        out[(size_t)row * FEATS + col] = o;
    }
}

// ------------------------------------------------- final 257 -> 1 layer (one wave/node)
__global__ void __launch_bounds__(256)
final_layer_kernel(const float* __restrict__ h,
                   const float* __restrict__ hagg,
                   const float* __restrict__ hea,
                   const float* __restrict__ W2,   // [257]
                   const float* __restrict__ b2,   // [1]
                   float* __restrict__ out) {      // [nodes]
    int gtid = blockIdx.x * blockDim.x + threadIdx.x;
    int n    = gtid >> 5;
    int lane = gtid & 31;
    if (n >= N_NODES) return;

    float4 x1 = *reinterpret_cast<const float4*>(h    + (size_t)n * FEATS + lane * 4);
    float4 w1 = *reinterpret_cast<const float4*>(W2   + lane * 4);
    float4 x2 = *reinterpret_cast<const float4*>(hagg + (size_t)n * FEATS + lane * 4);
    float4 w2 = *reinterpret_cast<const float4*>(W2   + 128 + lane * 4);

    float s = x1.x * w1.x + x1.y * w1.y + x1.z * w1.z + x1.w * w1.w
            + x2.x * w2.x + x2.y * w2.y + x2.z * w2.z + x2.w * w2.w;
#pragma unroll
    for (int m = 16; m >= 1; m >>= 1) s += __shfl_xor(s, m, 32);
    if (lane == 0) out[n] = s + hea[n] * W2[256] + b2[0];
}

// ----------------------------------------------------------------- driver
extern "C" void kernel_launch(void* const* d_in, const int* in_sizes, int n_in,
                              void* d_out, int out_size, void* d_ws, size_t ws_size,
                              hipStream_t stream) {
    const float* node_feat = (const float*)d_in[0];
    const float* edge_feat = (const float*)d_in[1];
    const int*   src       = (const int*)d_in[2];
    const int*   dst       = (const int*)d_in[3];
    const float* W1        = (const float*)d_in[4];
    const float* b1        = (const float*)d_in[5];
    const float* Wmid      = (const float*)d_in[6];  // [8][257][128]
    const float* bmid      = (const float*)d_in[7];  // [8][128]
    const float* W2        = (const float*)d_in[8];  // [257]
    const float* b2        = (const float*)d_in[9];  // [1]
    float*       out       = (float*)d_out;

    // workspace layout (floats), all 16B aligned
    float* ws      = (float*)d_ws;
    float* he_aggr = ws;                                   // N_NODES
    float* hA      = ws + N_NODES;                         // N*128
    float* hB      = hA + (size_t)N_NODES * FEATS;         // N*128
    float* hagg    = hB + (size_t)N_NODES * FEATS;         // N*128

    const int nmat4  = (N_NODES * FEATS) / 4;  // 3,200,000
    const int nvec4  = N_NODES / 4;            // 25,000
    const int zblkM  = (nmat4 + 255) / 256;
    const int zblkV  = (nvec4 + 255) / 256;
    const int eblk   = (N_EDGES + 255) / 256;        // edge-parallel, 1 thread/edge
    const int eblkW  = ((N_EDGES * 32) + 255) / 256; // wave-per-edge
    const int gblk   = N_NODES / 16;                 // 6250, exact

    // he_aggr is invariant across layers: compute once
    fill0_kernel<<<zblkV, 256, 0, stream>>>((float4*)he_aggr, nvec4);
    edge_aggr_e_kernel<<<eblk, 256, 0, stream>>>(edge_feat, dst, he_aggr);

    // 9 layers producing 128 features. act: layer 5 (Wmid[4]) and 8 (Wmid[7]) sigmoid
    const int acts[9] = {0, 0, 0, 0, 0, 1, 0, 0, 1};
    const float* h_in = node_feat;
    float*       h_out = hA;
    for (int L = 0; L < 9; ++L) {
        const float* W = (L == 0) ? W1 : (Wmid + (size_t)(L - 1) * 257 * FEATS);
        const float* b = (L == 0) ? b1 : (bmid + (size_t)(L - 1) * FEATS);
        fill0_kernel<<<zblkM, 256, 0, stream>>>((float4*)hagg, nmat4);
        edge_aggr_feat_kernel<<<eblkW, 256, 0, stream>>>(h_in, src, dst, hagg);
        sage_gemm_wmma_kernel<<<gblk, 256, 0, stream>>>(h_in, hagg, he_aggr, W, b,
                                                        h_out, acts[L]);
        h_in  = h_out;
        h_out = (h_out == hA) ? hB : hA;
    }

    // final 257 -> 1 layer
    fill0_kernel<<<zblkM, 256, 0, stream>>>((float4*)hagg, nmat4);
    edge_aggr_feat_kernel<<<eblkW, 256, 0, stream>>>(h_in, src, dst, hagg);
    final_layer_kernel<<<(N_NODES * 32 + 255) / 256, 256, 0, stream>>>(
        h_in, hagg, he_aggr, W2, b2, out);
}